// HeteroGNN_9775345565891
// MI455X (gfx1250) — compile-verified
//
#include <hip/hip_runtime.h>

typedef __attribute__((ext_vector_type(2))) float v2f;
typedef __attribute__((ext_vector_type(8))) float v8f;

#define D 128
#define OUTD 32
#define LDS_STRIDE 132   // 128 + 4 pad: row stride = 4 banks -> conflict-free frag reads
#define NEG_SLOPE 0.01f

// -------------------- native fp32 global atomic add (no CAS loop) --------------------
__device__ __forceinline__ void atomic_add_f32(float* p, float v) {
    asm volatile("global_atomic_add_f32 %0, %1, off" : : "v"(p), "v"(v) : "memory");
}

// -------------------- zero workspace region --------------------
__global__ void zero_kernel(float* __restrict__ p, long n) {
    long i = (long)blockIdx.x * blockDim.x + threadIdx.x;
    long stride = (long)gridDim.x * blockDim.x;
    for (; i < n; i += stride) p[i] = 0.0f;
}

// -------------------- pack weight into WMMA B-fragment order --------------------
// Fragment for k-step kk, lane-half h uses W rows 4kk+2h and 4kk+2h+1 = 2q, 2q+1 (q = 2kk+h).
// P[(q*ncols+col)*2 + {0,1}] = W0[2q][col] (+W1), W0[2q+1][col] (+W1)
// -> in the GEMM loop each lane does ONE aligned global_load_b64, contiguous across the half-wave.
__global__ void pack_weight_kernel(const float* __restrict__ W0, const float* __restrict__ W1,
                                   float* __restrict__ P, int ncols) {
    int idx = blockIdx.x * blockDim.x + threadIdx.x;
    int total = (D / 2) * ncols;                  // 64 row-pairs
    if (idx >= total) return;
    int q = idx / ncols;
    int col = idx - q * ncols;
    float a = W0[(2 * q) * ncols + col];
    float b = W0[(2 * q + 1) * ncols + col];
    if (W1 != nullptr) {
        a += W1[(2 * q) * ncols + col];
        b += W1[(2 * q + 1) * ncols + col];
    }
    P[(size_t)(q * ncols + col) * 2 + 0] = a;
    P[(size_t)(q * ncols + col) * 2 + 1] = b;
}

// -------------------- edge scatter: sum source features + degree count --------------------
// 32 lanes per edge; lane j handles features [4j, 4j+4). edge layout: [2, n_edges] (src row, dst row).
__global__ __launch_bounds__(256) void scatter_kernel(
    const float* __restrict__ x_src, const int* __restrict__ edge,
    float* __restrict__ agg, float* __restrict__ cnt, int n_edges)
{
    long t = (long)blockIdx.x * blockDim.x + threadIdx.x;
    int e = (int)(t >> 5);
    int chunk = (int)(t & 31);
    if (e >= n_edges) return;
    int src = edge[e];
    int dst = edge[n_edges + e];
    const float4 v = *(const float4*)(x_src + (size_t)src * D + chunk * 4);
    float* base = agg + (size_t)dst * D + chunk * 4;
    atomic_add_f32(base + 0, v.x);
    atomic_add_f32(base + 1, v.y);
    atomic_add_f32(base + 2, v.z);
    atomic_add_f32(base + 3, v.w);
    if (chunk == 0) atomic_add_f32(cnt + dst, 1.0f);
}

// -------------------- fused SAGEConv GEMM --------------------
// out[16 x 128] tile = (A0/cnt0)@Wl0 + bl0 [+ (A1/cnt1)@Wl1 + bl1] + Ar@Wr ; leaky relu.
// Weights arrive pre-packed in fragment order (pWr already contains Wr0+Wr2 for the paper path).
// 256 threads = 8 waves; wave w computes cols [16w, 16w+16) with v_wmma_f32_16x16x4_f32.
template <bool HAS1>
__global__ __launch_bounds__(256) void sage_gemm_kernel(
    const float* __restrict__ A0, const float* __restrict__ cnt0,
    const float* __restrict__ A1, const float* __restrict__ cnt1,
    const float* __restrict__ Ar,
    const float* __restrict__ pWl0, const float* __restrict__ bl0,
    const float* __restrict__ pWl1, const float* __restrict__ bl1,
    const float* __restrict__ pWr,
    float* __restrict__ out)
{
    __shared__ float sA0[16 * LDS_STRIDE];
    __shared__ float sA1[16 * LDS_STRIDE];
    __shared__ float sAr[16 * LDS_STRIDE];

    const int tid = threadIdx.x;
    const int r0  = blockIdx.x * 16;

    // stage A tiles: 2048 floats per matrix / 256 threads = 8 floats (2x float4) each
    {
        const int e   = tid * 8;
        const int row = e >> 7;          // tid / 16
        const int col = e & 127;
        const size_t goff = (size_t)(r0 + row) * D + col;
        const int loff = row * LDS_STRIDE + col;

        const float s0 = 1.0f / fmaxf(cnt0[r0 + row], 1.0f);
        float4 u0 = *(const float4*)(A0 + goff);
        float4 u1 = *(const float4*)(A0 + goff + 4);
        u0.x *= s0; u0.y *= s0; u0.z *= s0; u0.w *= s0;
        u1.x *= s0; u1.y *= s0; u1.z *= s0; u1.w *= s0;
        *(float4*)&sA0[loff]     = u0;
        *(float4*)&sA0[loff + 4] = u1;

        float4 r0v = *(const float4*)(Ar + goff);
        float4 r1v = *(const float4*)(Ar + goff + 4);
        *(float4*)&sAr[loff]     = r0v;
        *(float4*)&sAr[loff + 4] = r1v;

        if (HAS1) {
            const float s1 = 1.0f / fmaxf(cnt1[r0 + row], 1.0f);
            float4 w0 = *(const float4*)(A1 + goff);
            float4 w1 = *(const float4*)(A1 + goff + 4);
            w0.x *= s1; w0.y *= s1; w0.z *= s1; w0.w *= s1;
            w1.x *= s1; w1.y *= s1; w1.z *= s1; w1.w *= s1;
            *(float4*)&sA1[loff]     = w0;
            *(float4*)&sA1[loff + 4] = w1;
        }
    }
    __syncthreads();

    const int wave = tid >> 5;
    const int lane = tid & 31;
    const int half = lane >> 4;       // selects K pair (A cols / packed B row-pair)
    const int l15  = lane & 15;
    const int col  = wave * 16 + l15; // output column this lane contributes to

    // two accumulators (even/odd k-step) break the WMMA C-dependency chain
    v8f acc0 = {0.f, 0.f, 0.f, 0.f, 0.f, 0.f, 0.f, 0.f};
    v8f acc1 = {0.f, 0.f, 0.f, 0.f, 0.f, 0.f, 0.f, 0.f};

#pragma unroll
    for (int kk = 0; kk < D / 4; ++kk) {
        const int kA = kk * 4 + 2 * half;          // A-fragment column base in LDS
        const size_t bq = ((size_t)(kk * 2 + half) * D + col) * 2;  // packed B offset

        v8f& acc = (kk & 1) ? acc1 : acc0;

        v2f a0 = *(const v2f*)&sA0[l15 * LDS_STRIDE + kA];
        v2f b0 = *(const v2f*)(pWl0 + bq);
        acc = __builtin_amdgcn_wmma_f32_16x16x4_f32(false, a0, false, b0,
                                                    (short)0, acc, false, false);
        if (HAS1) {
            v2f a1 = *(const v2f*)&sA1[l15 * LDS_STRIDE + kA];
            v2f b1 = *(const v2f*)(pWl1 + bq);
            acc = __builtin_amdgcn_wmma_f32_16x16x4_f32(false, a1, false, b1,
                                                        (short)0, acc, false, false);
        }
        v2f ar = *(const v2f*)&sAr[l15 * LDS_STRIDE + kA];
        v2f br = *(const v2f*)(pWr + bq);
        acc = __builtin_amdgcn_wmma_f32_16x16x4_f32(false, ar, false, br,
                                                    (short)0, acc, false, false);
    }

    float bias = bl0[col];
    if (HAS1) bias += bl1[col];

    // C layout: lanes 0-15: VGPR v -> row v; lanes 16-31: VGPR v -> row v+8; col = lane&15 (+c0)
#pragma unroll
    for (int v = 0; v < 8; ++v) {
        float val = acc0[v] + acc1[v] + bias;
        val = val > 0.0f ? val : NEG_SLOPE * val;
        out[(size_t)(r0 + v + 8 * half) * D + col] = val;
    }
}

// -------------------- final projection: [N x 128] @ [128 x 32] + b (packed W) --------------------
__global__ __launch_bounds__(64) void final_linear_kernel(
    const float* __restrict__ A, const float* __restrict__ pW,
    const float* __restrict__ b, float* __restrict__ out)
{
    __shared__ float sA[16 * LDS_STRIDE];
    const int tid = threadIdx.x;
    const int r0  = blockIdx.x * 16;

    // stage 2048 floats with 64 threads: 4 iters x 8 floats
#pragma unroll
    for (int i = 0; i < 4; ++i) {
        const int e    = (tid + i * 64) * 8;
        const int row  = e >> 7;
        const int colx = e & 127;
        const float* g = A + (size_t)(r0 + row) * D + colx;
        *(float4*)&sA[row * LDS_STRIDE + colx]     = *(const float4*)g;
        *(float4*)&sA[row * LDS_STRIDE + colx + 4] = *(const float4*)(g + 4);
    }
    __syncthreads();

    const int wave = tid >> 5;
    const int lane = tid & 31;
    const int half = lane >> 4;
    const int l15  = lane & 15;
    const int col  = wave * 16 + l15;

    v8f acc0 = {0.f, 0.f, 0.f, 0.f, 0.f, 0.f, 0.f, 0.f};
    v8f acc1 = {0.f, 0.f, 0.f, 0.f, 0.f, 0.f, 0.f, 0.f};
#pragma unroll
    for (int kk = 0; kk < D / 4; ++kk) {
        const int kA = kk * 4 + 2 * half;
        const size_t bq = ((size_t)(kk * 2 + half) * OUTD + col) * 2;
        v8f& acc = (kk & 1) ? acc1 : acc0;
        v2f a  = *(const v2f*)&sA[l15 * LDS_STRIDE + kA];
        v2f bb = *(const v2f*)(pW + bq);
        acc = __builtin_amdgcn_wmma_f32_16x16x4_f32(false, a, false, bb,
                                                    (short)0, acc, false, false);
    }
    const float bias = b[col];
#pragma unroll
    for (int v = 0; v < 8; ++v)
        out[(size_t)(r0 + v + 8 * half) * OUTD + col] = acc0[v] + acc1[v] + bias;
}

// -------------------- host orchestration --------------------
extern "C" void kernel_launch(void* const* d_in, const int* in_sizes, int n_in,
                              void* d_out, int out_size, void* d_ws, size_t ws_size,
                              hipStream_t stream) {
    const float* x_author = (const float*)d_in[0];
    const float* x_paper  = (const float*)d_in[1];
    const int* e_writes   = (const int*)d_in[2];
    const int* e_writtenby= (const int*)d_in[3];
    const int* e_cites    = (const int*)d_in[4];
    const float* Wl       = (const float*)d_in[5];   // [2,3,128,128]
    const float* bl       = (const float*)d_in[6];   // [2,3,128]
    const float* Wr       = (const float*)d_in[7];   // [2,3,128,128]
    const float* lin_W    = (const float*)d_in[8];   // [128,32]
    const float* lin_b    = (const float*)d_in[9];   // [32]

    const int n_author = in_sizes[0] / D;            // 50000
    const int n_paper  = in_sizes[1] / D;            // 50000
    const int ne       = in_sizes[2] / 2;            // 800000

    const size_t NF  = (size_t)n_paper * D;          // node-feature slab
    const size_t WSZ = (size_t)D * D;                // weight slab (16384)
    float* ws   = (float*)d_ws;
    float* agg1 = ws;                  // writes  -> paper
    float* agg2 = agg1 + NF;           // cites   -> paper
    float* agg3 = agg2 + NF;           // written -> author
    float* cnt1 = agg3 + NF;
    float* cnt2 = cnt1 + n_paper;
    float* cnt3 = cnt2 + n_paper;
    float* abuf0 = cnt3 + n_author;
    float* pbuf0 = abuf0 + NF;
    float* abuf1 = pbuf0 + NF;
    float* pbuf1 = abuf1 + NF;
    // packed weights: per layer 5 slabs (pWl0, pWl2, pWrP(=Wr0+Wr2), pWl1, pWrA) + packed lin_W
    float* pk    = pbuf1 + NF;
    float* pLin  = pk + 2 * 5 * WSZ;

    // ---- pack all weights up front (tiny) ----
    {
        const int pb = (int)((WSZ + 255) / 256);     // 64 blocks for 128x128
        for (int l = 0; l < 2; ++l) {
            const float* Wl0 = Wl + ((size_t)l * 3 + 0) * WSZ;
            const float* Wl1 = Wl + ((size_t)l * 3 + 1) * WSZ;
            const float* Wl2 = Wl + ((size_t)l * 3 + 2) * WSZ;
            const float* Wr0 = Wr + ((size_t)l * 3 + 0) * WSZ;
            const float* Wr1 = Wr + ((size_t)l * 3 + 1) * WSZ;
            const float* Wr2 = Wr + ((size_t)l * 3 + 2) * WSZ;
            float* base = pk + (size_t)l * 5 * WSZ;
            pack_weight_kernel<<<pb, 256, 0, stream>>>(Wl0, nullptr, base + 0 * WSZ, D);
            pack_weight_kernel<<<pb, 256, 0, stream>>>(Wl2, nullptr, base + 1 * WSZ, D);
            pack_weight_kernel<<<pb, 256, 0, stream>>>(Wr0, Wr2,     base + 2 * WSZ, D); // pre-summed root W
            pack_weight_kernel<<<pb, 256, 0, stream>>>(Wl1, nullptr, base + 3 * WSZ, D);
            pack_weight_kernel<<<pb, 256, 0, stream>>>(Wr1, nullptr, base + 4 * WSZ, D);
        }
        const int lb = (int)(((size_t)(D / 2) * OUTD + 255) / 256);
        pack_weight_kernel<<<lb, 256, 0, stream>>>(lin_W, nullptr, pLin, OUTD);
    }

    const long zero_n = 3 * (long)NF + 2 * n_paper + n_author;
    const int scat_blocks = (int)(((long)ne * 32 + 255) / 256);
    const int ptiles = n_paper / 16, atiles = n_author / 16;

    const float* a_in = x_author;
    const float* p_in = x_paper;
    float* a_out = abuf0;
    float* p_out = pbuf0;

    for (int l = 0; l < 2; ++l) {
        zero_kernel<<<2048, 256, 0, stream>>>(agg1, zero_n);

        scatter_kernel<<<scat_blocks, 256, 0, stream>>>(a_in, e_writes,    agg1, cnt1, ne);
        scatter_kernel<<<scat_blocks, 256, 0, stream>>>(p_in, e_cites,     agg2, cnt2, ne);
        scatter_kernel<<<scat_blocks, 256, 0, stream>>>(p_in, e_writtenby, agg3, cnt3, ne);

        float* base = pk + (size_t)l * 5 * WSZ;
        // papers: relations 0 (writes) and 2 (cites)
        sage_gemm_kernel<true><<<ptiles, 256, 0, stream>>>(
            agg1, cnt1, agg2, cnt2, p_in,
            base + 0 * WSZ, bl + ((size_t)l * 3 + 0) * D,
            base + 1 * WSZ, bl + ((size_t)l * 3 + 2) * D,
            base + 2 * WSZ,
            p_out);
        // authors: relation 1 (written_by)
        sage_gemm_kernel<false><<<atiles, 256, 0, stream>>>(
            agg3, cnt3, nullptr, nullptr, a_in,
            base + 3 * WSZ, bl + ((size_t)l * 3 + 1) * D,
            nullptr, nullptr,
            base + 4 * WSZ,
            a_out);

        a_in = a_out; p_in = p_out;
        a_out = abuf1; p_out = pbuf1;
    }

    final_linear_kernel<<<atiles, 64, 0, stream>>>(a_in, pLin, lin_b, (float*)d_out);
}